// VectorQuantizer_47777216201281
// MI455X (gfx1250) — compile-verified
//
#include <hip/hip_runtime.h>

typedef __attribute__((ext_vector_type(2))) float v2f;
typedef __attribute__((ext_vector_type(4))) float v4f;
typedef __attribute__((ext_vector_type(8))) float v8f;

#define EMBED_SIZE 8   // K: codebook entries
#define EMBED_DIM  4   // D: embedding dim

// One wave processes 2 tiles of 16 latent points per iteration, one
// V_WMMA_F32_16X16X4_F32 per tile.
//   A (16x4)  = -2 * codebook, rows 8..15 zero padded  (wave-invariant)
//   B (4x16)  = 16 latent points as columns
//   C (16x16) = ||c_m||^2 broadcast along columns (0 for pad rows)
//   D[m][n]   = ||c_m||^2 - 2 * x_n . c_m   (same argmin as true L2 dist)
// D layout: lane n (0..15) holds codes m=0..7 in its 8 accumulator regs ->
// in-lane argmin, then gather codebook row, NT-store to both outputs.
// No private arrays with divergent indices -> zero LDS, max occupancy.
__global__ void vq_wmma_kernel(const float* __restrict__ latent,
                               const float* __restrict__ codebook,
                               float* __restrict__ out,
                               long long npoints, long long nwaves) {
  const int lane = threadIdx.x & 31;
  const int half = lane >> 4;   // 0: dims {0,1}; 1: dims {2,3}
  const int col  = lane & 15;   // point-in-tile / code row
  const long long wave =
      (long long)blockIdx.x * (blockDim.x >> 5) + (threadIdx.x >> 5);

  // ---- A matrix: -2 * codebook, 16x4, per ISA 32-bit A layout ----
  // lane L: M = col; VGPR0 holds K = 2*half, VGPR1 holds K = 2*half+1.
  // Direct strided load from the (cache-resident) codebook; zero pad rows.
  {
    // declared below, before cbias, to keep scopes tidy
  }
  const int ccol = col & (EMBED_SIZE - 1);
  v2f arow = *(const v2f*)(codebook + ccol * EMBED_DIM + 2 * half);
  const float ascale = (col < EMBED_SIZE) ? -2.0f : 0.0f;
  v2f a;
  a[0] = ascale * arow[0];
  a[1] = ascale * arow[1];

  // ---- C matrix: row bias ||c_m||^2 (VGPR r -> M=r lanes 0-15; M=r+8 pad) ----
  // Constant indices only -> stays in VGPRs.
  v8f cbias;
  #pragma unroll
  for (int r = 0; r < EMBED_SIZE; ++r) {
    v4f row = *(const v4f*)(codebook + r * EMBED_DIM);
    float n2 = row[0]*row[0] + row[1]*row[1] + row[2]*row[2] + row[3]*row[3];
    cbias[r] = (half == 0) ? n2 : 0.0f;
  }

  const long long T = (npoints + 15) >> 4;          // tiles of 16 points
  const long long outStride = npoints * EMBED_DIM;  // floats between outputs
  const long long step = nwaves * 2;

  for (long long t = wave * 2; t < T; t += step) {
    const long long p0 = t * 16 + col;
    const long long p1 = p0 + 16;
    const long long pl0 = (p0 < npoints) ? p0 : (npoints - 1);
    const long long pl1 = (p1 < npoints) ? p1 : (npoints - 1);

    // B matrices: lane L loads dims {2*half, 2*half+1} of point (L&15).
    // Two back-to-back coalesced NT b64 loads (512 B contiguous per wave).
    v2f b0 = __builtin_nontemporal_load(
        (const v2f*)(latent + pl0 * EMBED_DIM + 2 * half));
    v2f b1 = __builtin_nontemporal_load(
        (const v2f*)(latent + pl1 * EMBED_DIM + 2 * half));

    // D = A*B + C  (emits v_wmma_f32_16x16x4_f32)
    v8f d0 = __builtin_amdgcn_wmma_f32_16x16x4_f32(
        false, a, false, b0, (short)0, cbias, false, false);
    v8f d1 = __builtin_amdgcn_wmma_f32_16x16x4_f32(
        false, a, false, b1, (short)0, cbias, false, false);

    // In-lane argmin over 8 codes (strict < => first-min, matches jnp.argmin)
    float best0 = d0[0], best1 = d1[0];
    int bi0 = 0, bi1 = 0;
    #pragma unroll
    for (int k = 1; k < EMBED_SIZE; ++k) {
      if (d0[k] < best0) { best0 = d0[k]; bi0 = k; }
      if (d1[k] < best1) { best1 = d1[k]; bi1 = k; }
    }

    // Gather selected code rows (cache hits).
    v4f q0 = *(const v4f*)(codebook + bi0 * EMBED_DIM);
    v4f q1 = *(const v4f*)(codebook + bi1 * EMBED_DIM);

    if (half == 0) {
      if (p0 < npoints) {
        float* o = out + p0 * EMBED_DIM;
        __builtin_nontemporal_store(q0, (v4f*)o);               // policy_vq
        __builtin_nontemporal_store(q0, (v4f*)(o + outStride)); // quantized
      }
      if (p1 < npoints) {
        float* o = out + p1 * EMBED_DIM;
        __builtin_nontemporal_store(q1, (v4f*)o);
        __builtin_nontemporal_store(q1, (v4f*)(o + outStride));
      }
    }
    // if-block re-converges: EXEC is all-1s again before the next WMMA
  }
}

extern "C" void kernel_launch(void* const* d_in, const int* in_sizes, int n_in,
                              void* d_out, int out_size, void* d_ws, size_t ws_size,
                              hipStream_t stream) {
  const float* latent   = (const float*)d_in[0];
  const float* codebook = (const float*)d_in[1];
  float* out = (float*)d_out;

  const long long nfloats = (long long)in_sizes[0];     // N * D
  const long long npoints = nfloats / EMBED_DIM;

  const int threads = 256;                 // 8 waves per block (wave32)
  const int wavesPerBlock = threads / 32;
  const int blocks = 2048;                 // 16384 waves, ~8 dual-tile iters each
  const long long nwaves = (long long)blocks * wavesPerBlock;

  vq_wmma_kernel<<<blocks, threads, 0, stream>>>(latent, codebook, out,
                                                 npoints, nwaves);
}